// EquivariantProductBasisBlock_31233002176563
// MI455X (gfx1250) — compile-verified
//
#include <hip/hip_runtime.h>

typedef __attribute__((ext_vector_type(2))) float v2f;
typedef __attribute__((ext_vector_type(8))) float v8f;

#define NN 1024     // nodes
#define CH 128      // channels
#define NE 10       // species
#define DD 9        // irreps dim
#define NP3 8
#define NP2 3
#define OUT_COLS 1152
#define KSTEPS 19   // K = 72 (U3: p*9+k) + 3 (U2: p) + 1 pad = 76 = 19*4
#define TSTRIDE (KSTEPS * 32 * 2)   // floats per packed 16-col tile

// per m-block (m = 1,3,5): columns m*81 = 81/243/405 ; 16-wide col tiles
// 6/16/26 (48 total).

struct G1Params {
    const float* U1[3];
    const float* W1[3];
    const float* W2[3];
    const float* W3[3];
};

// ---------------------------------------------------------------- species ----
__global__ __launch_bounds__(256) void species_kernel(const float* __restrict__ attrs,
                                                      int* __restrict__ sp) {
    int b = blockIdx.x * blockDim.x + threadIdx.x;
    if (b >= NN) return;
    float best = -1.0f; int bi = 0;
    #pragma unroll
    for (int e = 0; e < NE; ++e) {
        float v = attrs[b * NE + e];
        if (v > best) { best = v; bi = e; }
    }
    sp[b] = bi;
}

// ------------------------------------------------------------------- pack ----
// Pack the fused B matrix (U3 rows 0..71, U2 rows 72..74, zero row 75) into
// the WMMA B-fragment layout for V_WMMA_F32_16X16X4_F32:
// lanes 0-15 hold N=lane, K={k0,k0+1}; lanes 16-31 hold N=lane-16, K={k0+2,k0+3}.
// Col q = mc*81 + i*9 + j, zero-padded to the 16-col tile boundary.
__global__ __launch_bounds__(256) void pack_kernel(const float* __restrict__ U3_0,
                                                   const float* __restrict__ U3_1,
                                                   const float* __restrict__ U3_2,
                                                   const float* __restrict__ U2_0,
                                                   const float* __restrict__ U2_1,
                                                   const float* __restrict__ U2_2,
                                                   float* __restrict__ Bpack) {
    int tid = blockIdx.x * blockDim.x + threadIdx.x;
    if (tid >= 48 * KSTEPS * 32) return;
    int lane = tid & 31;
    int ks   = (tid >> 5) % KSTEPS;
    int tile = tid / (KSTEPS * 32);      // global col-tile 0..47
    int g, tloc;
    if (tile < 6)       { g = 0; tloc = tile; }
    else if (tile < 22) { g = 1; tloc = tile - 6; }
    else                { g = 2; tloc = tile - 22; }
    const float* U3 = (g == 0) ? U3_0 : ((g == 1) ? U3_1 : U3_2);
    const float* U2 = (g == 0) ? U2_0 : ((g == 1) ? U2_1 : U2_2);
    const int ncg = (g == 0) ? 81 : ((g == 1) ? 243 : 405);
    int rsel = (lane < 16) ? 0 : 2;
    int col  = tloc * 16 + (lane & 15);
    float v0 = 0.0f, v1 = 0.0f;
    if (col < ncg) {
        int mc  = col / 81;
        int rem = col - mc * 81;
        int i   = rem / 9;
        int j   = rem - i * 9;
        int r0  = ks * 4 + rsel;
        int r1  = r0 + 1;
        if (r0 < 72) {
            int p0 = r0 / 9, k0 = r0 - p0 * 9;
            v0 = U3[(((mc * DD + i) * DD + j) * DD + k0) * NP3 + p0];
        } else if (r0 < 75) {
            v0 = U2[((mc * DD + i) * DD + j) * NP2 + (r0 - 72)];
        }
        if (r1 < 72) {
            int p1 = r1 / 9, k1 = r1 - p1 * 9;
            v1 = U3[(((mc * DD + i) * DD + j) * DD + k1) * NP3 + p1];
        } else if (r1 < 75) {
            v1 = U2[((mc * DD + i) * DD + j) * NP2 + (r1 - 72)];
        }
    }
    Bpack[tid * 2 + 0] = v0;
    Bpack[tid * 2 + 1] = v1;
}

// --------------------------------------------------------------------- G1 ----
// One wave per 16-row tile of the (b,c) = N*C row space (rows share node b,
// hence species).  The fused WMMA GEMM produces t = U3:(w3⊗x) + U2·w2 per
// column (mc,i,j); the VALU epilogue does only the two x-contractions and the
// U1 term, writing out1 as 9 stacked [N x C] matrices.
struct WaveSmem {
    float xs[16][10];       // x[row][k], padded
    float w3[16][8];
    float w2[16][4];
    float w1[16];
    float t3[16][416];      // largest padded col count (26 tiles * 16)
};

__global__ __launch_bounds__(64) void g1_kernel(const float* __restrict__ x,
                                                const int* __restrict__ species,
                                                const float* __restrict__ Bpack,
                                                G1Params P,
                                                float* __restrict__ Obuf) {
    __shared__ WaveSmem sm[2];
    const int lane   = threadIdx.x & 31;
    const int waveId = threadIdx.x >> 5;
    const int tile   = blockIdx.x * 2 + waveId;   // 0..8191
    const int row0   = tile * 16;                 // rows row0..row0+15
    const int b      = row0 >> 7;                 // node index
    const int c0     = row0 & 127;                // channel base
    const int e      = species[b];
    WaveSmem& S = sm[waveId];

    for (int idx = lane; idx < 16 * 9; idx += 32) {
        int r = idx / 9, k = idx - r * 9;
        S.xs[r][k] = x[(row0 + r) * DD + k];
    }
    __syncthreads();

    const int MG[3]   = {1, 3, 5};
    const int TIL[3]  = {6, 16, 26};
    const int TOFF[3] = {0, 6, 22};
    const int OOFF[3] = {0, 1, 4};

    const int rsel = (lane < 16) ? 0 : 2;
    const int rl   = lane & 15;

    #pragma unroll
    for (int g = 0; g < 3; ++g) {
        const int m = MG[g];
        for (int idx = lane; idx < 16 * 8; idx += 32) {
            int r = idx >> 3, p = idx & 7;
            S.w3[r][p] = P.W3[g][(e * NP3 + p) * CH + (c0 + r)];
        }
        for (int idx = lane; idx < 16 * 3; idx += 32) {
            int r = idx / 3, p = idx - r * 3;
            S.w2[r][p] = P.W2[g][(e * NP2 + p) * CH + (c0 + r)];
        }
        if (lane < 16) S.w1[lane] = P.W1[g][e * CH + (c0 + lane)];
        __syncthreads();

        // A fragments, shared by every column tile of this m-block.
        // Rows 0..71: w3[p]*x[k]; rows 72..74: w2[p]; row 75: 0.
        v2f afrag[KSTEPS];
        #pragma unroll
        for (int ks = 0; ks < 18; ++ks) {
            int r0 = ks * 4 + rsel;
            int p0 = r0 / 9, k0 = r0 - p0 * 9;
            int r1 = r0 + 1;
            int p1 = r1 / 9, k1 = r1 - p1 * 9;
            v2f a;
            a.x = S.w3[rl][p0] * S.xs[rl][k0];
            a.y = S.w3[rl][p1] * S.xs[rl][k1];
            afrag[ks] = a;
        }
        {
            v2f a;                       // rows 72..75 (the U2 k-step)
            if (lane < 16) { a.x = S.w2[rl][0]; a.y = S.w2[rl][1]; }
            else           { a.x = S.w2[rl][2]; a.y = 0.0f; }
            afrag[18] = a;
        }

        // fused GEMM over column tiles: t = A(16x76) * Bpack(76 x 16*tiles).
        // Bulk-load all 19 B fragments into distinct registers first so the
        // global_load_b64s stay in flight together instead of serializing
        // each WMMA behind a full-latency load.
        for (int ct = 0; ct < TIL[g]; ++ct) {
            const v2f* Bt =
                (const v2f*)(Bpack + (size_t)(TOFF[g] + ct) * TSTRIDE);
            v2f bfrag[KSTEPS];
            #pragma unroll
            for (int ks = 0; ks < KSTEPS; ++ks)
                bfrag[ks] = Bt[ks * 32 + lane];
            v8f acc = {0.f, 0.f, 0.f, 0.f, 0.f, 0.f, 0.f, 0.f};
            #pragma unroll
            for (int ks = 0; ks < KSTEPS; ++ks)
                acc = __builtin_amdgcn_wmma_f32_16x16x4_f32(
                    false, afrag[ks], false, bfrag[ks], (short)0, acc,
                    false, false);
            const int mBase = (lane < 16) ? 0 : 8;
            #pragma unroll
            for (int rr = 0; rr < 8; ++rr)
                S.t3[mBase + rr][ct * 16 + rl] = acc[rr];
        }
        __syncthreads();

        // epilogue: out2[mc,i] = sum_j t[mc,i,j] x[j]
        //           out1[mc]   = sum_i (U1[mc,i] w1 + out2[mc,i]) x[i]
        const float* U1g = P.U1[g];
        for (int pair = lane; pair < 16 * m; pair += 32) {
            int r  = pair / m;
            int mc = pair - r * m;
            float acc1 = 0.0f;
            #pragma unroll
            for (int i = 0; i < 9; ++i) {
                float acc2 = 0.0f;
                #pragma unroll
                for (int j = 0; j < 9; ++j)
                    acc2 += S.t3[r][mc * 81 + i * 9 + j] * S.xs[r][j];
                float u1w1 = U1g[mc * DD + i] * S.w1[r];
                acc1 += (u1w1 + acc2) * S.xs[r][i];
            }
            Obuf[((size_t)(OOFF[g] + mc) * NN + b) * CH + (c0 + r)] = acc1;
        }
        __syncthreads();
    }
}

// --------------------------------------------------------------------- G2 ----
// Equivariant linear per (g, mc): seg[b,v] = (1/sqrt(C)) * sum_u O[b,u] wl[u,v]
// as a WMMA GEMM [1024 x 128] * [128 x 128], fused with +sc and the
// interleaved (v*m + mc) output layout.  Fragments are preloaded in chunks of
// 8 k-steps to keep loads in flight across the WMMA chain.
__global__ __launch_bounds__(128) void g2_kernel(const float* __restrict__ Obuf,
                                                 const float* __restrict__ wl0,
                                                 const float* __restrict__ wl1,
                                                 const float* __restrict__ wl2,
                                                 const float* __restrict__ sc,
                                                 float* __restrict__ out) {
    const int lane = threadIdx.x & 31;
    const int wave = threadIdx.x >> 5;
    const int job  = blockIdx.x * 4 + wave;  // < 9 * 64 * 8 = 4608
    const int nt   = job & 7;
    const int mt   = (job >> 3) & 63;
    const int mcg  = job >> 9;               // 0..8 (stacked (g,mc))
    const int g    = (mcg == 0) ? 0 : ((mcg < 4) ? 1 : 2);
    const int mc   = mcg - ((g == 0) ? 0 : ((g == 1) ? 1 : 4));
    const int m    = (g == 0) ? 1 : ((g == 1) ? 3 : 5);
    const int coff = (g == 0) ? 0 : ((g == 1) ? 128 : 512);
    const float* wl = (g == 0) ? wl0 : ((g == 1) ? wl1 : wl2);
    const float* A  = Obuf + (size_t)mcg * NN * CH;

    const int rl   = lane & 15;
    const int ksel = (lane < 16) ? 0 : 2;
    const int brow = mt * 16 + rl;   // A fragment M index
    const int vcol = nt * 16 + rl;   // B fragment N index

    v8f acc = {0.f, 0.f, 0.f, 0.f, 0.f, 0.f, 0.f, 0.f};
    #pragma unroll
    for (int kk = 0; kk < 4; ++kk) {
        v2f  af[8];
        float bx[8], by[8];
        #pragma unroll
        for (int q = 0; q < 8; ++q) {
            int k0 = (kk * 8 + q) * 4 + ksel;
            af[q] = *(const v2f*)&A[(size_t)brow * CH + k0];
            bx[q] = wl[(k0) * CH + vcol];
            by[q] = wl[(k0 + 1) * CH + vcol];
        }
        #pragma unroll
        for (int q = 0; q < 8; ++q) {
            v2f bb; bb.x = bx[q]; bb.y = by[q];
            acc = __builtin_amdgcn_wmma_f32_16x16x4_f32(
                false, af[q], false, bb, (short)0, acc, false, false);
        }
    }

    const float s = 0.08838834764831845f;  // 1/sqrt(128)
    const int mBase = (lane < 16) ? 0 : 8;
    #pragma unroll
    for (int rr = 0; rr < 8; ++rr) {
        int bnode = mt * 16 + mBase + rr;
        int vv    = nt * 16 + (lane & 15);
        int col   = coff + vv * m + mc;
        size_t idx = (size_t)bnode * OUT_COLS + col;
        out[idx] = acc[rr] * s + sc[idx];
    }
}

// ------------------------------------------------------------------ launch ---
extern "C" void kernel_launch(void* const* d_in, const int* in_sizes, int n_in,
                              void* d_out, int out_size, void* d_ws, size_t ws_size,
                              hipStream_t stream) {
    (void)in_sizes; (void)n_in; (void)out_size; (void)ws_size;
    const float* node_feats = (const float*)d_in[0];
    const float* node_attrs = (const float*)d_in[1];
    const float* sc         = (const float*)d_in[2];

    const float *U3p[3], *U2p[3], *wlp[3];
    G1Params gp;
    for (int g = 0; g < 3; ++g) {
        gp.U1[g] = (const float*)d_in[3 + 3 * g + 0];
        U2p[g]   = (const float*)d_in[3 + 3 * g + 1];
        U3p[g]   = (const float*)d_in[3 + 3 * g + 2];
        gp.W1[g] = (const float*)d_in[12 + 3 * g + 0];
        gp.W2[g] = (const float*)d_in[12 + 3 * g + 1];
        gp.W3[g] = (const float*)d_in[12 + 3 * g + 2];
        wlp[g]   = (const float*)d_in[21 + g];
    }

    // workspace (floats): [species:1024 ints][Bpack:48*TSTRIDE][Obuf:9*N*C]
    int*   species = (int*)d_ws;
    float* Bpack   = (float*)d_ws + 1024;
    float* Obuf    = Bpack + 48 * TSTRIDE;

    species_kernel<<<4, 256, 0, stream>>>(node_attrs, species);
    pack_kernel<<<(48 * KSTEPS * 32 + 255) / 256, 256, 0, stream>>>(
        U3p[0], U3p[1], U3p[2], U2p[0], U2p[1], U2p[2], Bpack);
    g1_kernel<<<4096, 64, 0, stream>>>(node_feats, species, Bpack, gp, Obuf);
    g2_kernel<<<1152, 128, 0, stream>>>(Obuf, wlp[0], wlp[1], wlp[2], sc,
                                        (float*)d_out);
}